// GraphTransformerExtractor_66365834658164
// MI455X (gfx1250) — compile-verified
//
#include <hip/hip_runtime.h>
#include <math.h>

// ---------------------------------------------------------------------------
// GraphTransformer VAE extractor for MI455X (gfx1250), fp32 WMMA pipeline.
// GEMMs register-block NT column-tiles per wave so each A (activation) fetch
// feeds NT v_wmma_f32_16x16x4_f32 ops; B fetches are batched ahead of the
// WMMA group so loads overlap the matrix pipe (staggered s_wait_loadcnt).
// ---------------------------------------------------------------------------

typedef float v2f __attribute__((ext_vector_type(2)));
typedef float v8f __attribute__((ext_vector_type(8)));

#define N_NODES   50000
#define N_EDGES   400000
#define IN_DIM_   128
#define NGRAPHS   50

// ---- monotonic float <-> uint transform for atomicMax-based segment max ----
__device__ __forceinline__ unsigned f2ord(float f) {
    unsigned u = __float_as_uint(f);
    return (u & 0x80000000u) ? ~u : (u | 0x80000000u);
}
__device__ __forceinline__ float ord2f(unsigned u) {
    u = (u & 0x80000000u) ? (u & 0x7fffffffu) : ~u;
    return __uint_as_float(u);
}

// ---------------------------------------------------------------------------
// GEMM: out[M][Nc] = X[M][K] @ W[Nc][K]^T + bias, via V_WMMA_F32_16X16X4_F32.
// One wave computes a 16 x (16*NT) strip; K stepped by 4.
//   A tile 16x4 : lane<16 -> row=lane, K={k,k+1}; lane>=16 -> K={k+2,k+3}
//   B tile 4x16 : lane<16 -> col=lane, K={k,k+1}; lane>=16 -> K={k+2,k+3}
//                 (B[k][n] = W[n][k] -> contiguous float2 from row-major W)
//   C/D         : VGPR g -> M = g + (lane>=16 ? 8 : 0), N = lane & 15
// Requires M % 16 == 0, Nc % (16*NT) == 0, K % 4 == 0.
// ---------------------------------------------------------------------------
template <int NT>
__global__ void gemm_xwt_bias(const float* __restrict__ X,
                              const float* __restrict__ W,
                              const float* __restrict__ bias,
                              float* __restrict__ out,
                              int M, int K, int Nc)
{
    int wave = blockIdx.x * (blockDim.x >> 5) + (threadIdx.x >> 5);
    int lane = threadIdx.x & 31;
    int ngroups = Nc / (16 * NT);
    int mt = wave / ngroups;
    int ng = wave - mt * ngroups;
    if (mt * 16 >= M) return;              // wave-uniform exit: EXEC stays all-1s

    int row  = mt * 16 + (lane & 15);
    int koff = (lane >> 4) * 2;            // upper half-wave owns K+2,K+3
    const float* xp = X + (size_t)row * K + koff;

    const float* wp[NT];
    int cols[NT];
    #pragma unroll
    for (int t = 0; t < NT; ++t) {
        cols[t] = (ng * NT + t) * 16 + (lane & 15);
        wp[t]   = W + (size_t)cols[t] * K + koff;
    }

    v8f acc[NT];
    #pragma unroll
    for (int t = 0; t < NT; ++t) acc[t] = (v8f){};

    for (int k = 0; k < K; k += 4) {
        // Batch all fetches for this K-step first so the scheduler keeps
        // NT+1 loads in flight and drains them with partial loadcnt waits
        // while earlier WMMAs execute.
        v2f a = *(const v2f*)(xp + k);
        v2f b[NT];
        #pragma unroll
        for (int t = 0; t < NT; ++t) b[t] = *(const v2f*)(wp[t] + k);
        #pragma unroll
        for (int t = 0; t < NT; ++t)
            acc[t] = __builtin_amdgcn_wmma_f32_16x16x4_f32(
                         /*neg_a=*/false, a, /*neg_b=*/false, b[t],
                         /*c_mod=*/(short)0, acc[t],
                         /*reuse_a=*/false, /*reuse_b=*/false);
    }

    int rbase = mt * 16 + ((lane >> 4) << 3);
    #pragma unroll
    for (int t = 0; t < NT; ++t) {
        float bv = bias[cols[t]];
        #pragma unroll
        for (int g = 0; g < 8; ++g)
            out[(size_t)(rbase + g) * Nc + cols[t]] = acc[t][g] + bv;
    }
}

// ---------------------------------------------------------------------------
// Edge attention passes
// ---------------------------------------------------------------------------
__global__ void edge_logit(const float* __restrict__ q, const float* __restrict__ k,
                           const int* __restrict__ src, const int* __restrict__ dst,
                           float* __restrict__ wbuf, unsigned* __restrict__ mbuf,
                           int E, int H, int C, float scale)
{
    int idx = blockIdx.x * blockDim.x + threadIdx.x;   // over E*H
    if (idx >= E * H) return;
    int e = idx / H, h = idx - e * H;
    int s = src[e], d = dst[e];
    const float* qp = q + (size_t)d * (H * C) + h * C;
    const float* kp = k + (size_t)s * (H * C) + h * C;
    float acc = 0.f;
    for (int c = 0; c < C; ++c) acc += qp[c] * kp[c];
    acc *= scale;
    wbuf[idx] = acc;
    atomicMax(mbuf + (size_t)d * H + h, f2ord(acc));
}

__global__ void edge_exp(const int* __restrict__ dst, const unsigned* __restrict__ mbuf,
                         float* __restrict__ wbuf, float* __restrict__ denom, int E, int H)
{
    int idx = blockIdx.x * blockDim.x + threadIdx.x;
    if (idx >= E * H) return;
    int e = idx / H, h = idx - e * H;
    int d = dst[e];
    float w = __expf(wbuf[idx] - ord2f(mbuf[(size_t)d * H + h]));
    wbuf[idx] = w;
    atomicAdd(denom + (size_t)d * H + h, w);
}

__global__ void edge_norm(const int* __restrict__ dst, const float* __restrict__ denom,
                          float* __restrict__ wbuf, int E, int H)
{
    int idx = blockIdx.x * blockDim.x + threadIdx.x;
    if (idx >= E * H) return;
    int e = idx / H, h = idx - e * H;
    wbuf[idx] /= (denom[(size_t)dst[e] * H + h] + 1e-16f);
}

// agg was pre-filled with the skip-GEMM result; scatter alpha * v on top.
__global__ void edge_scatter(const int* __restrict__ src, const int* __restrict__ dst,
                             const float* __restrict__ wbuf, const float* __restrict__ v,
                             float* __restrict__ agg, int E, int H, int C)
{
    int F = H * C;
    int idx = blockIdx.x * blockDim.x + threadIdx.x;   // over E*F (<= 204.8M)
    if (idx >= E * F) return;
    int e = idx / F, j = idx - e * F, h = j / C;
    float alpha = wbuf[e * H + h];
    atomicAdd(agg + (size_t)dst[e] * F + j, alpha * v[(size_t)src[e] * F + j]);
}

// ---------------------------------------------------------------------------
// Elementwise / pooling helpers
// ---------------------------------------------------------------------------
__global__ void fill_u32(unsigned* __restrict__ p, unsigned v, long n)
{
    long i = (long)blockIdx.x * blockDim.x + threadIdx.x;
    if (i < n) p[i] = v;
}

__global__ void relu_ip(float* __restrict__ x, long n)
{
    long i = (long)blockIdx.x * blockDim.x + threadIdx.x;
    if (i < n) x[i] = fmaxf(x[i], 0.f);
}

__global__ void vae_out(const float* __restrict__ mu, const float* __restrict__ logvar,
                        const float* __restrict__ eps, const float* __restrict__ dru,
                        const int* __restrict__ batch,
                        float* __restrict__ out, float* __restrict__ pooled,
                        int n, int dim)
{
    int idx = blockIdx.x * blockDim.x + threadIdx.x;
    if (idx >= n * dim) return;
    float z = mu[idx] + eps[idx] * __expf(0.5f * logvar[idx]);
    float o = (dru[idx] >= 0.5f) ? z * 2.0f : 0.0f;     // Dropout(p=0.5), train
    out[idx] = o;
    int node = idx / dim;
    atomicAdd(pooled + (size_t)batch[node] * dim + (idx - node * dim), o);
}

__global__ void count_nodes(const int* __restrict__ batch, float* __restrict__ cnt, int n)
{
    int i = blockIdx.x * blockDim.x + threadIdx.x;
    if (i < n) atomicAdd(cnt + batch[i], 1.0f);
}

__global__ void pool_div(float* __restrict__ pooled, const float* __restrict__ cnt,
                         int g, int dim)
{
    int i = blockIdx.x * blockDim.x + threadIdx.x;
    if (i < g * dim) pooled[i] /= fmaxf(cnt[i / dim], 1.0f);
}

// ---------------------------------------------------------------------------
// Host side
// ---------------------------------------------------------------------------
static inline int cdiv_l(long a, int b) { return (int)((a + b - 1) / b); }

static inline void launch_gemm(const float* X, const float* W, const float* b,
                               float* out, int M, int K, int Nc, hipStream_t s)
{
    if (Nc % 64 == 0) {
        int tiles  = (M / 16) * (Nc / 64);             // 16x64 strip per wave
        int blocks = (tiles + 7) / 8;                  // 8 waves (256 thr) / block
        hipLaunchKernelGGL((gemm_xwt_bias<4>), dim3(blocks), dim3(256), 0, s,
                           X, W, b, out, M, K, Nc);
    } else {                                           // Nc == 32 (input proj)
        int tiles  = (M / 16) * (Nc / 32);             // 16x32 strip per wave
        int blocks = (tiles + 7) / 8;
        hipLaunchKernelGGL((gemm_xwt_bias<2>), dim3(blocks), dim3(256), 0, s,
                           X, W, b, out, M, K, Nc);
    }
}

extern "C" void kernel_launch(void* const* d_in, const int* in_sizes, int n_in,
                              void* d_out, int out_size, void* d_ws, size_t ws_size,
                              hipStream_t stream)
{
    (void)in_sizes; (void)n_in; (void)out_size; (void)ws_size;

    // ---- inputs (setup_inputs() dict order, params flattened depth-first) ----
    const float* x     = (const float*)d_in[0];
    const int*   ei    = (const int*)  d_in[1];
    const int*   batch = (const int*)  d_in[2];
    const float* eps   = (const float*)d_in[3];
    const float* dru   = (const float*)d_in[4];
    const float* ipw   = (const float*)d_in[5];
    const float* ipb   = (const float*)d_in[6];
    const float* muw   = (const float*)d_in[7];
    const float* mub   = (const float*)d_in[8];
    const float* lvw   = (const float*)d_in[9];
    const float* lvb   = (const float*)d_in[10];
    // layers[li]: q.w q.b k.w k.b v.w v.b skip.w skip.b  -> base 11 + 8*li
    const int*   src   = ei;              // edge_index[0]
    const int*   dst   = ei + N_EDGES;    // edge_index[1]

    // ---- workspace carve-up (floats) ----
    float* wsf = (float*)d_ws;
    size_t o = 0;
    const size_t NB = (size_t)N_NODES * 512;
    float* h0   = wsf + o; o += NB;       // ping
    float* h1   = wsf + o; o += NB;       // pong / agg
    float* qb   = wsf + o; o += NB;
    float* kb   = wsf + o; o += NB;
    float* vb   = wsf + o; o += NB;
    float* wbuf = wsf + o; o += (size_t)N_EDGES * 4;     // per-(edge,head) weights
    unsigned* mbuf = (unsigned*)(wsf + o); o += (size_t)N_NODES * 4;
    float* dbuf = wsf + o; o += (size_t)N_NODES * 4;
    float* cnt  = wsf + o; o += 64;

    float* outp   = (float*)d_out;                       // [N, 128]
    float* pooled = outp + (size_t)N_NODES * IN_DIM_;    // [50, 128]

    const int DIMS[5]  = {32, 64, 128, 256, 512};
    const int HEADS[4] = {4, 4, 2, 1};

    // ---- input projection: h0 = x @ ipw^T + ipb ----
    launch_gemm(x, ipw, ipb, h0, N_NODES, IN_DIM_, DIMS[0], stream);

    float* h[2] = {h0, h1};
    int cur = 0;
    for (int li = 0; li < 4; ++li) {
        int fin  = DIMS[li], fout = DIMS[li + 1];
        int H    = HEADS[li], C = fout / H;
        const float* qw = (const float*)d_in[11 + 8 * li + 0];
        const float* qc = (const float*)d_in[11 + 8 * li + 1];
        const float* kw = (const float*)d_in[11 + 8 * li + 2];
        const float* kc = (const float*)d_in[11 + 8 * li + 3];
        const float* vw = (const float*)d_in[11 + 8 * li + 4];
        const float* vc = (const float*)d_in[11 + 8 * li + 5];
        const float* sw = (const float*)d_in[11 + 8 * li + 6];
        const float* sc = (const float*)d_in[11 + 8 * li + 7];
        float* hin = h[cur];
        float* agg = h[cur ^ 1];          // skip GEMM result doubles as accumulator

        launch_gemm(hin, qw, qc, qb,  N_NODES, fin, fout, stream);
        launch_gemm(hin, kw, kc, kb,  N_NODES, fin, fout, stream);
        launch_gemm(hin, vw, vc, vb,  N_NODES, fin, fout, stream);
        launch_gemm(hin, sw, sc, agg, N_NODES, fin, fout, stream);

        long nh = (long)N_NODES * H;
        hipLaunchKernelGGL(fill_u32, dim3(cdiv_l(nh, 256)), dim3(256), 0, stream,
                           mbuf, 0u, nh);                       // ordered -inf
        hipLaunchKernelGGL(fill_u32, dim3(cdiv_l(nh, 256)), dim3(256), 0, stream,
                           (unsigned*)dbuf, 0u, nh);            // 0.0f

        int eh = N_EDGES * H;
        float scale = 1.0f / sqrtf((float)C);
        hipLaunchKernelGGL(edge_logit, dim3(cdiv_l(eh, 256)), dim3(256), 0, stream,
                           qb, kb, src, dst, wbuf, mbuf, N_EDGES, H, C, scale);
        hipLaunchKernelGGL(edge_exp, dim3(cdiv_l(eh, 256)), dim3(256), 0, stream,
                           dst, mbuf, wbuf, dbuf, N_EDGES, H);
        hipLaunchKernelGGL(edge_norm, dim3(cdiv_l(eh, 256)), dim3(256), 0, stream,
                           dst, dbuf, wbuf, N_EDGES, H);
        long ef = (long)N_EDGES * fout;
        hipLaunchKernelGGL(edge_scatter, dim3(cdiv_l(ef, 256)), dim3(256), 0, stream,
                           src, dst, wbuf, vb, agg, N_EDGES, H, C);
        long nf = (long)N_NODES * fout;
        hipLaunchKernelGGL(relu_ip, dim3(cdiv_l(nf, 256)), dim3(256), 0, stream,
                           agg, nf);
        cur ^= 1;
    }

    // ---- VAE head: mu -> qb, logvar -> kb ----
    launch_gemm(h[cur], muw, mub, qb, N_NODES, DIMS[4], IN_DIM_, stream);
    launch_gemm(h[cur], lvw, lvb, kb, N_NODES, DIMS[4], IN_DIM_, stream);

    long pg = (long)NGRAPHS * IN_DIM_;
    hipLaunchKernelGGL(fill_u32, dim3(cdiv_l(pg, 256)), dim3(256), 0, stream,
                       (unsigned*)pooled, 0u, pg);
    hipLaunchKernelGGL(fill_u32, dim3(1), dim3(64), 0, stream,
                       (unsigned*)cnt, 0u, (long)NGRAPHS);

    long nd = (long)N_NODES * IN_DIM_;
    hipLaunchKernelGGL(vae_out, dim3(cdiv_l(nd, 256)), dim3(256), 0, stream,
                       qb, kb, eps, dru, batch, outp, pooled, N_NODES, IN_DIM_);
    hipLaunchKernelGGL(count_nodes, dim3(cdiv_l(N_NODES, 256)), dim3(256), 0, stream,
                       batch, cnt, N_NODES);
    hipLaunchKernelGGL(pool_div, dim3(cdiv_l(pg, 256)), dim3(256), 0, stream,
                       pooled, cnt, NGRAPHS, IN_DIM_);
}